// DyHGCN_10685878632723
// MI455X (gfx1250) — compile-verified
//
#include <hip/hip_runtime.h>
#include <hip/hip_bf16.h>

#define USER_NUM 50000
#define D_MODEL  64
#define POS_DIM  8
#define N_HEADS  8
#define D_K      64
#define T_STEPS  8
#define N_EDGES  400000
#define BSZ      16
#define SEQ      201
#define LSEQ     200            // SEQ - 1
#define IN_SIZE  72             // D_MODEL + POS_DIM
#define QKV_DIM  512            // N_HEADS * D_K
#define NEG_INF  (-4294967295.0f)  // -(2^32 - 1)

typedef float v2f __attribute__((ext_vector_type(2)));
typedef float v8f __attribute__((ext_vector_type(8)));

// ---------------------------------------------------------------------------
// Software-pipelined K loop: fragments for step kk+4 are loaded BEFORE the
// WMMAs of step kk issue, so each WMMA waits on loads issued one full
// iteration earlier (latency hidden inside the wave, not only across waves).
// MASKED=false is the wave-uniform fast path (no clamp/mask ALU at all).
// K must be a multiple of 4 and >= 8 (64 / 72 / 512 here).
// ---------------------------------------------------------------------------
template<int MT, bool BT, bool MASKED>
__device__ __forceinline__ void wmma_kloop_f32(
    const float* __restrict__ bp, long bstep, int ldb,
    const float* __restrict__ (&ap)[MT], const float (&amsk)[MT],
    float bmsk, int K, v8f (&c)[MT])
{
    // ---- prologue: load K-step 0 fragments --------------------------------
    v2f bf, af[MT];
    if (BT) { bf.x = bp[0]; bf.y = bp[1]; }
    else    { bf.x = bp[0]; bf.y = bp[ldb]; }
#pragma unroll
    for (int m = 0; m < MT; ++m) { af[m].x = ap[m][0]; af[m].y = ap[m][1]; }

    // ---- steady state: prefetch kk+4, compute kk --------------------------
#pragma unroll 2
    for (int kk = 0; kk < K - 4; kk += 4) {
        bp += bstep;
        v2f bn;
        if (BT) { bn.x = bp[0]; bn.y = bp[1]; }
        else    { bn.x = bp[0]; bn.y = bp[ldb]; }
        v2f an[MT];
#pragma unroll
        for (int m = 0; m < MT; ++m) {
            an[m].x = ap[m][4];
            an[m].y = ap[m][5];
            ap[m] += 4;
        }

        v2f bb = bf;
        if (MASKED) { bb.x *= bmsk; bb.y *= bmsk; }
#pragma unroll
        for (int m = 0; m < MT; ++m) {
            v2f aa = af[m];
            if (MASKED) { aa.x *= amsk[m]; aa.y *= amsk[m]; }
            c[m] = __builtin_amdgcn_wmma_f32_16x16x4_f32(
                       false, aa, false, bb, (short)0, c[m], false, false);
        }

        bf = bn;
#pragma unroll
        for (int m = 0; m < MT; ++m) af[m] = an[m];
    }

    // ---- epilogue: last K-step --------------------------------------------
    v2f bb = bf;
    if (MASKED) { bb.x *= bmsk; bb.y *= bmsk; }
#pragma unroll
    for (int m = 0; m < MT; ++m) {
        v2f aa = af[m];
        if (MASKED) { aa.x *= amsk[m]; aa.y *= amsk[m]; }
        c[m] = __builtin_amdgcn_wmma_f32_16x16x4_f32(
                   false, aa, false, bb, (short)0, c[m], false, false);
    }
}

// ---------------------------------------------------------------------------
// fp32 WMMA GEMM: C[M x N] = A[M x K] @ B (+bias) (+relu)
// BT == false : B is [K x N] row-major, ldb = row stride
// BT == true  : B memory is [N x K] row-major (computes A @ B^T), ldb = row stride
// Each wave computes a (16*MT) x 16 macro-tile of C via V_WMMA_F32_16X16X4_F32,
// reusing one B fragment across MT accumulators per K-step.
// Full interior tiles (wave-uniform test) run a branch-free unmasked loop with
// unguarded stores; edge tiles use clamped addresses + 0/1 mask multiplies so
// EXEC stays all-ones at every WMMA (ISA requirement).
// ---------------------------------------------------------------------------
template<int MT, bool BT>
__global__ void wmma_gemm_f32(const float* __restrict__ A,
                              const float* __restrict__ Bm,
                              const float* __restrict__ bias,
                              float* __restrict__ C,
                              int M, int N, int K,
                              int lda, int ldb, long ldc,
                              int doRelu)
{
    const int lane = threadIdx.x;        // 0..31 (wave32)
    const int lo   = lane & 15;
    const int hi   = lane >> 4;          // 0 or 1

    const int col0 = (blockIdx.x * blockDim.y + threadIdx.y) * 16;
    const int row0 = blockIdx.y * (16 * MT);
    if (col0 >= N) return;               // wave-uniform exit

    const bool full = (row0 + 16 * MT <= M) && (col0 + 16 <= N);  // wave-uniform

    v8f c[MT];
#pragma unroll
    for (int m = 0; m < MT; ++m) c[m] = (v8f){};

    const int col = col0 + lo;

    if (full) {
        // ---------------- fast path: no masks, no guards ----------------
        const float* bp;
        long bstep;
        if (BT) { bp = Bm + (long)col * ldb + 2 * hi;  bstep = 4; }
        else    { bp = Bm + (long)(2 * hi) * ldb + col; bstep = 4L * ldb; }

        const float* ap[MT];
        float amsk[MT];
#pragma unroll
        for (int m = 0; m < MT; ++m) {
            ap[m]   = A + (long)(row0 + m * 16 + lo) * lda + 2 * hi;
            amsk[m] = 1.0f;
        }

        wmma_kloop_f32<MT, BT, false>(bp, bstep, ldb, ap, amsk, 1.0f, K, c);

        const float bv = bias ? bias[col] : 0.0f;
#pragma unroll
        for (int m = 0; m < MT; ++m) {
#pragma unroll
            for (int r = 0; r < 8; ++r) {
                const int row = row0 + m * 16 + r + 8 * hi;   // ISA C-frag row map
                float v = c[m][r] + bv;
                if (doRelu) v = fmaxf(v, 0.0f);
                C[(long)row * ldc + col] = v;
            }
        }
    } else {
        // ---------------- edge path: clamped + masked -------------------
        const int  cB    = col < N ? col : (N - 1);
        const float bmsk = (col < N) ? 1.0f : 0.0f;
        const float* bp;
        long bstep;
        if (BT) { bp = Bm + (long)cB * ldb + 2 * hi;  bstep = 4; }
        else    { bp = Bm + (long)(2 * hi) * ldb + cB; bstep = 4L * ldb; }

        const float* ap[MT];
        float amsk[MT];
#pragma unroll
        for (int m = 0; m < MT; ++m) {
            int r   = row0 + m * 16 + lo;
            amsk[m] = (r < M) ? 1.0f : 0.0f;
            int rc  = r < M ? r : (M - 1);
            ap[m]   = A + (long)rc * lda + 2 * hi;
        }

        wmma_kloop_f32<MT, BT, true>(bp, bstep, ldb, ap, amsk, bmsk, K, c);

        if (col < N) {
            const float bv = bias ? bias[col] : 0.0f;
#pragma unroll
            for (int m = 0; m < MT; ++m) {
#pragma unroll
                for (int r = 0; r < 8; ++r) {
                    const int row = row0 + m * 16 + r + 8 * hi;
                    if (row < M) {
                        float v = c[m][r] + bv;
                        if (doRelu) v = fmaxf(v, 0.0f);
                        C[(long)row * ldc + col] = v;
                    }
                }
            }
        }
    }
}

// ---------------------------------------------------------------------------
// GCN: degree init (self-loop => 1.0)
// ---------------------------------------------------------------------------
__global__ void k_deg_init(float* __restrict__ deg)
{
    int i = blockIdx.x * blockDim.x + threadIdx.x;
    if (i < T_STEPS * USER_NUM) deg[i] = 1.0f;
}

__global__ void k_deg_count(const int* __restrict__ edge_index,
                            float* __restrict__ deg)
{
    int e = blockIdx.x * blockDim.x + threadIdx.x;
    int t = blockIdx.y;
    if (e < N_EDGES) {
        int dst = edge_index[((long)t * 2 + 1) * N_EDGES + e];
        atomicAdd(&deg[t * USER_NUM + dst], 1.0f);
    }
}

// node[t][u][d] = gcn_b[d] + xw[u][d] / max(deg,1)   (self-loop term)
__global__ void k_node_init(const float* __restrict__ xw,
                            const float* __restrict__ deg,
                            const float* __restrict__ gcn_b,
                            float* __restrict__ node)
{
    int d = threadIdx.x;   // blockDim.x == 64
    int u = blockIdx.x;
    int t = blockIdx.y;
    float dg = fmaxf(deg[t * USER_NUM + u], 1.0f);
    node[((long)t * USER_NUM + u) * D_MODEL + d] =
        gcn_b[d] + xw[(long)u * D_MODEL + d] / dg;
}

// scatter: node[t][dst] += xw[src] * rsqrt(deg[src]) * rsqrt(deg[dst])
// one wave per edge, 2 dims per lane
__global__ void k_gcn_scatter(const int* __restrict__ edge_index,
                              const float* __restrict__ xw,
                              const float* __restrict__ deg,
                              float* __restrict__ node)
{
    int lane   = threadIdx.x & 31;
    int eLocal = threadIdx.x >> 5;
    long e = (long)blockIdx.x * (blockDim.x >> 5) + eLocal;
    int t = blockIdx.y;
    if (e >= N_EDGES) return;
    int s = edge_index[((long)t * 2)     * N_EDGES + e];
    int d = edge_index[((long)t * 2 + 1) * N_EDGES + e];
    float w = rsqrtf(fmaxf(deg[t * USER_NUM + s], 1.0f)) *
              rsqrtf(fmaxf(deg[t * USER_NUM + d], 1.0f));
    const float* xs = xw + (long)s * D_MODEL;
    float* nd = node + ((long)t * USER_NUM + d) * D_MODEL;
    atomicAdd(&nd[lane],      xs[lane]      * w);
    atomicAdd(&nd[lane + 32], xs[lane + 32] * w);
}

// ---------------------------------------------------------------------------
// Time attention over T_STEPS + concat pos_emb -> final[3200 x 72]
// one thread per (b,l)
// ---------------------------------------------------------------------------
__global__ void k_dyemb(const int* __restrict__ input_seq,
                        const int* __restrict__ ts_idx,
                        const float* __restrict__ time_emb,
                        const float* __restrict__ node,
                        const float* __restrict__ pos_emb,
                        float* __restrict__ final_)
{
    int row = blockIdx.x * blockDim.x + threadIdx.x;
    if (row >= BSZ * LSEQ) return;
    int b = row / LSEQ, l = row % LSEQ;
    int u  = input_seq[b * SEQ + l];
    int ts = ts_idx[b * LSEQ + l];
    const float* tq = time_emb + (long)ts * D_MODEL;

    float sc[T_STEPS];
    float mx = -1e30f;
    for (int t = 0; t < T_STEPS; ++t) {
        const float* nv = node + ((long)t * USER_NUM + u) * D_MODEL;
        float acc = 0.0f;
        for (int d = 0; d < D_MODEL; ++d) acc += tq[d] * nv[d];
        sc[t] = acc * 0.125f;   // / sqrt(64)
        mx = fmaxf(mx, sc[t]);
    }
    float sum = 0.0f;
    for (int t = 0; t < T_STEPS; ++t) { sc[t] = __expf(sc[t] - mx); sum += sc[t]; }
    float inv = 1.0f / sum;

    float* out = final_ + (long)row * IN_SIZE;
    for (int d = 0; d < D_MODEL; ++d) {
        float acc = 0.0f;
        for (int t = 0; t < T_STEPS; ++t)
            acc += sc[t] * node[((long)t * USER_NUM + u) * D_MODEL + d];
        out[d] = acc * inv;
    }
    for (int j = 0; j < POS_DIM; ++j) out[D_MODEL + j] = pos_emb[l * POS_DIM + j];
}

// ---------------------------------------------------------------------------
// Masked causal self-attention, one block per (q, h, b)
// ---------------------------------------------------------------------------
__global__ void k_attn(const float* __restrict__ Q,
                       const float* __restrict__ K,
                       const float* __restrict__ V,
                       const int* __restrict__ input_seq,
                       float* __restrict__ Vatt)
{
    __shared__ float sc[256];
    __shared__ float red[2];
    int q = blockIdx.x, h = blockIdx.y, b = blockIdx.z;
    int tid = threadIdx.x;
    bool qpad = (input_seq[b * SEQ + q] == 0);
    const float scale = 1.0f / (8.0f + 1e-6f);

    if (tid < LSEQ) {
        float v;
        if (tid > q || qpad) {
            v = NEG_INF;
        } else {
            const float* qp = Q + (long)(b * LSEQ + q)   * QKV_DIM + h * D_K;
            const float* kp = K + (long)(b * LSEQ + tid) * QKV_DIM + h * D_K;
            float acc = 0.0f;
            for (int d = 0; d < D_K; ++d) acc += qp[d] * kp[d];
            v = acc * scale;
        }
        sc[tid] = v;
    }
    __syncthreads();
    if (tid == 0) {
        float mx = sc[0];
        for (int i = 1; i < LSEQ; ++i) mx = fmaxf(mx, sc[i]);
        red[0] = mx;
    }
    __syncthreads();
    float mx = red[0];
    if (tid < LSEQ) sc[tid] = __expf(sc[tid] - mx);
    __syncthreads();
    if (tid == 0) {
        float s = 0.0f;
        for (int i = 0; i < LSEQ; ++i) s += sc[i];
        red[1] = s;
    }
    __syncthreads();
    float inv = 1.0f / red[1];
    if (tid < D_K) {
        float acc = 0.0f;
        for (int k = 0; k < LSEQ; ++k)
            acc += sc[k] * V[(long)(b * LSEQ + k) * QKV_DIM + h * D_K + tid];
        Vatt[(long)(b * LSEQ + q) * QKV_DIM + h * D_K + tid] = acc * inv;
    }
}

// ---------------------------------------------------------------------------
// out = LayerNorm(x + res) * g + beta   (rows of IN_SIZE=72)
// ---------------------------------------------------------------------------
__global__ void k_add_ln(const float* __restrict__ x,
                         const float* __restrict__ res,
                         const float* __restrict__ g,
                         const float* __restrict__ beta,
                         float* __restrict__ out, int rows)
{
    int row = blockIdx.x * blockDim.x + threadIdx.x;
    if (row >= rows) return;
    const float* xp = x   + (long)row * IN_SIZE;
    const float* rp = res + (long)row * IN_SIZE;
    float v[IN_SIZE];
    float mu = 0.0f;
    for (int i = 0; i < IN_SIZE; ++i) { v[i] = xp[i] + rp[i]; mu += v[i]; }
    mu *= (1.0f / IN_SIZE);
    float var = 0.0f;
    for (int i = 0; i < IN_SIZE; ++i) { float dd = v[i] - mu; var += dd * dd; }
    var *= (1.0f / IN_SIZE);
    float is = rsqrtf(var + 1e-5f);
    float* op = out + (long)row * IN_SIZE;
    for (int i = 0; i < IN_SIZE; ++i) op[i] = (v[i] - mu) * is * g[i] + beta[i];
}

// ---------------------------------------------------------------------------
// prev-mask: logits[row][u] -= 1000 for each distinct u in prefix inp[b,0..l],
// plus unconditionally at u == 0 (set semantics, dedup via first occurrence)
// ---------------------------------------------------------------------------
__global__ void k_prevmask(const int* __restrict__ input_seq,
                           float* __restrict__ out)
{
    int row = blockIdx.x;             // b*LSEQ + l
    int b = row / LSEQ, l = row % LSEQ;
    float* op = out + (long)row * USER_NUM;
    if (threadIdx.x == 0) op[0] -= 1000.0f;
    for (int k = threadIdx.x; k <= l; k += blockDim.x) {
        int u = input_seq[b * SEQ + k];
        if (u == 0) continue;
        bool first = true;
        for (int kp = 0; kp < k; ++kp)
            if (input_seq[b * SEQ + kp] == u) { first = false; break; }
        if (first) op[u] -= 1000.0f;
    }
}

// ---------------------------------------------------------------------------
extern "C" void kernel_launch(void* const* d_in, const int* in_sizes, int n_in,
                              void* d_out, int out_size, void* d_ws, size_t ws_size,
                              hipStream_t stream)
{
    (void)in_sizes; (void)n_in; (void)out_size; (void)ws_size;

    const float* user_emb = (const float*)d_in[0];
    const float* gcn_W    = (const float*)d_in[1];
    const float* gcn_b    = (const float*)d_in[2];
    const float* time_emb = (const float*)d_in[3];
    const float* pos_emb  = (const float*)d_in[4];
    const float* W_q      = (const float*)d_in[5];
    const float* W_k      = (const float*)d_in[6];
    const float* W_v      = (const float*)d_in[7];
    const float* W_o      = (const float*)d_in[8];
    const float* ffn_W1   = (const float*)d_in[9];
    const float* ffn_b1   = (const float*)d_in[10];
    const float* ffn_W2   = (const float*)d_in[11];
    const float* ffn_b2   = (const float*)d_in[12];
    const float* ln1_g    = (const float*)d_in[13];
    const float* ln1_b    = (const float*)d_in[14];
    const float* ln2_g    = (const float*)d_in[15];
    const float* ln2_b    = (const float*)d_in[16];
    const float* out_W    = (const float*)d_in[17];
    const float* out_b    = (const float*)d_in[18];
    const int*   input_seq = (const int*)d_in[19];
    const int*   edge_idx  = (const int*)d_in[20];
    const int*   dy_ts     = (const int*)d_in[21];

    float* logits = (float*)d_out;
    const int ROWS = BSZ * LSEQ;    // 3200

    // workspace layout (floats)
    float* ws = (float*)d_ws;
    float* xw    = ws;                                  // 50000*64
    float* deg   = xw    + (long)USER_NUM * D_MODEL;    // 8*50000
    float* node  = deg   + (long)T_STEPS * USER_NUM;    // 8*50000*64
    float* fin   = node  + (long)T_STEPS * USER_NUM * D_MODEL;  // 3200*72
    float* Qm    = fin   + (long)ROWS * IN_SIZE;        // 3200*512
    float* Km    = Qm    + (long)ROWS * QKV_DIM;
    float* Vm    = Km    + (long)ROWS * QKV_DIM;
    float* Vatt  = Vm    + (long)ROWS * QKV_DIM;
    float* Oproj = Vatt  + (long)ROWS * QKV_DIM;        // 3200*72
    float* Xbuf  = Oproj + (long)ROWS * IN_SIZE;
    float* Hbuf  = Xbuf  + (long)ROWS * IN_SIZE;
    float* Fbuf  = Hbuf  + (long)ROWS * IN_SIZE;
    float* attO  = Fbuf  + (long)ROWS * IN_SIZE;        // att_out, 3200*72

    dim3 wblk(32, 4);   // 4 waves/block; each wave: 64 rows x 16 cols of C

    // 1) xw = user_emb @ gcn_W   (50000x64 @ 64x64)
    wmma_gemm_f32<4, false><<<dim3(1, (USER_NUM + 63) / 64), wblk, 0, stream>>>(
        user_emb, gcn_W, nullptr, xw,
        USER_NUM, D_MODEL, D_MODEL, D_MODEL, D_MODEL, D_MODEL, 0);

    // 2) GCN per timestep
    k_deg_init<<<(T_STEPS * USER_NUM + 255) / 256, 256, 0, stream>>>(deg);
    k_deg_count<<<dim3((N_EDGES + 255) / 256, T_STEPS), 256, 0, stream>>>(edge_idx, deg);
    k_node_init<<<dim3(USER_NUM, T_STEPS), D_MODEL, 0, stream>>>(xw, deg, gcn_b, node);
    k_gcn_scatter<<<dim3(N_EDGES / 8, T_STEPS), 256, 0, stream>>>(edge_idx, xw, deg, node);

    // 3) time-attention + pos concat -> final
    k_dyemb<<<(ROWS + 127) / 128, 128, 0, stream>>>(
        input_seq, dy_ts, time_emb, node, pos_emb, fin);

    // 4) Q, K, V projections (3200x72 @ 72x512)
    wmma_gemm_f32<4, false><<<dim3(QKV_DIM / 64, ROWS / 64), wblk, 0, stream>>>(
        fin, W_q, nullptr, Qm, ROWS, QKV_DIM, IN_SIZE, IN_SIZE, QKV_DIM, QKV_DIM, 0);
    wmma_gemm_f32<4, false><<<dim3(QKV_DIM / 64, ROWS / 64), wblk, 0, stream>>>(
        fin, W_k, nullptr, Km, ROWS, QKV_DIM, IN_SIZE, IN_SIZE, QKV_DIM, QKV_DIM, 0);
    wmma_gemm_f32<4, false><<<dim3(QKV_DIM / 64, ROWS / 64), wblk, 0, stream>>>(
        fin, W_v, nullptr, Vm, ROWS, QKV_DIM, IN_SIZE, IN_SIZE, QKV_DIM, QKV_DIM, 0);

    // 5) masked causal attention
    k_attn<<<dim3(LSEQ, N_HEADS, BSZ), 256, 0, stream>>>(Qm, Km, Vm, input_seq, Vatt);

    // 6) O projection (3200x512 @ 512x72), then X = LN1(final + Vatt@Wo)
    wmma_gemm_f32<4, false><<<dim3(2, ROWS / 64), wblk, 0, stream>>>(
        Vatt, W_o, nullptr, Oproj, ROWS, IN_SIZE, QKV_DIM, QKV_DIM, IN_SIZE, IN_SIZE, 0);
    k_add_ln<<<(ROWS + 127) / 128, 128, 0, stream>>>(Oproj, fin, ln1_g, ln1_b, Xbuf, ROWS);

    // 7) FFN: H = relu(X@W1 + b1); F = H@W2 + b2; att_out = LN2(F + X)
    wmma_gemm_f32<4, false><<<dim3(2, ROWS / 64), wblk, 0, stream>>>(
        Xbuf, ffn_W1, ffn_b1, Hbuf, ROWS, IN_SIZE, IN_SIZE, IN_SIZE, IN_SIZE, IN_SIZE, 1);
    wmma_gemm_f32<4, false><<<dim3(2, ROWS / 64), wblk, 0, stream>>>(
        Hbuf, ffn_W2, ffn_b2, Fbuf, ROWS, IN_SIZE, IN_SIZE, IN_SIZE, IN_SIZE, IN_SIZE, 0);
    k_add_ln<<<(ROWS + 127) / 128, 128, 0, stream>>>(Fbuf, Xbuf, ln2_g, ln2_b, attO, ROWS);

    // 8) logits = att_out @ out_W^T + out_b   (3200x72 @ 72x50000, B transposed)
    wmma_gemm_f32<4, true><<<dim3((USER_NUM / 16 + 3) / 4, ROWS / 64), wblk, 0, stream>>>(
        attO, out_W, out_b, logits,
        ROWS, USER_NUM, IN_SIZE, IN_SIZE, IN_SIZE, (long)USER_NUM, 0);

    // 9) previously-seen-user mask (set semantics, -1000)
    k_prevmask<<<ROWS, 128, 0, stream>>>(input_seq, logits);
}